// LAM_Module_30399778521224
// MI455X (gfx1250) — compile-verified
//
#include <hip/hip_runtime.h>
#include <hip/hip_bf16.h>

typedef __attribute__((ext_vector_type(2))) float v2f;
typedef __attribute__((ext_vector_type(8))) float v8f;

#define D_TOT   1048576LL   // C*H*W = 64*128*128
#define N_ROWS  32
#define KPW     256         // k-positions per wave in gram pass

// ---------------------------------------------------------------------------
// Zero the Gram accumulator (2 * 32 * 32 floats).
// ---------------------------------------------------------------------------
__global__ void zero_kernel(float* __restrict__ G, int n) {
    for (int i = threadIdx.x; i < n; i += blockDim.x) G[i] = 0.0f;
}

// ---------------------------------------------------------------------------
// Pass 1: G[b] = q[b] * q[b]^T   (32x32 per batch), via V_WMMA_F32_16X16X4_F32.
// grid = (512, B), block = 256 (8 waves). Each wave owns KPW consecutive k.
// One float2 load per lane doubles as A-fragment (rows) and B-fragment (cols).
// ---------------------------------------------------------------------------
__global__ void gram_kernel(const float* __restrict__ x, float* __restrict__ G) {
    const int b    = blockIdx.y;
    const int lane = threadIdx.x & 31;
    const int wave = threadIdx.x >> 5;

    const long long k0 = ((long long)blockIdx.x * 8 + wave) * KPW;
    const float* q = x + (long long)b * N_ROWS * D_TOT;

    const int half = lane >> 4;      // 0: K={0,1}, 1: K={2,3} of each chunk
    const int lm   = lane & 15;      // row/col within tile
    const int kk   = half * 2;

    const float* rowA = q + (long long)lm * D_TOT;        // rows 0..15
    const float* rowB = q + (long long)(16 + lm) * D_TOT; // rows 16..31

    v8f c00 = {}, c01 = {}, c10 = {}, c11 = {};
    for (int it = 0; it < KPW / 4; ++it) {
        const long long kb = k0 + (long long)it * 4 + kk;
        v2f f0 = *(const v2f*)(rowA + kb);   // fragment for rows/cols 0..15
        v2f f1 = *(const v2f*)(rowB + kb);   // fragment for rows/cols 16..31
        c00 = __builtin_amdgcn_wmma_f32_16x16x4_f32(false, f0, false, f0, (short)0, c00, false, false);
        c01 = __builtin_amdgcn_wmma_f32_16x16x4_f32(false, f0, false, f1, (short)0, c01, false, false);
        c10 = __builtin_amdgcn_wmma_f32_16x16x4_f32(false, f1, false, f0, (short)0, c10, false, false);
        c11 = __builtin_amdgcn_wmma_f32_16x16x4_f32(false, f1, false, f1, (short)0, c11, false, false);
    }

    // Block-level reduction of the 32x32 partials in LDS, then one global
    // atomic sweep per block (keeps global atomic contention ~512/address).
    __shared__ float sG[N_ROWS * N_ROWS];
    for (int i = threadIdx.x; i < N_ROWS * N_ROWS; i += blockDim.x) sG[i] = 0.0f;
    __syncthreads();

    const int rbase = half * 8;      // C/D layout: lanes 16-31 hold M+8
    const int col   = lm;
#pragma unroll
    for (int r = 0; r < 8; ++r) {
        atomicAdd(&sG[(rbase + r)      * 32 + col],      c00[r]);
        atomicAdd(&sG[(rbase + r)      * 32 + 16 + col], c01[r]);
        atomicAdd(&sG[(16 + rbase + r) * 32 + col],      c10[r]);
        atomicAdd(&sG[(16 + rbase + r) * 32 + 16 + col], c11[r]);
    }
    __syncthreads();

    float* Gb = G + b * N_ROWS * N_ROWS;
    for (int i = threadIdx.x; i < N_ROWS * N_ROWS; i += blockDim.x)
        atomicAdd(&Gb[i], sG[i]);
}

// ---------------------------------------------------------------------------
// Middle: attention = gamma * softmax(rowmax(E) - E), E = normalized Gram.
// 1 block, 64 threads: thread -> (b, row).
// ---------------------------------------------------------------------------
__global__ void attn_kernel(const float* __restrict__ G,
                            const float* __restrict__ gamma,
                            float* __restrict__ A) {
    const int b = threadIdx.x >> 5;
    const int i = threadIdx.x & 31;
    const float* Gb = G + b * 1024;

    const float ni = fmaxf(sqrtf(Gb[i * 32 + i]), 1e-12f);
    float e[32];
    float rowmax = -3.0e38f;
#pragma unroll
    for (int j = 0; j < 32; ++j) {
        const float nj = fmaxf(sqrtf(Gb[j * 32 + j]), 1e-12f);
        const float v = Gb[i * 32 + j] / (ni * nj);
        e[j] = v;
        rowmax = fmaxf(rowmax, v);
    }
    float m2 = -3.0e38f;
#pragma unroll
    for (int j = 0; j < 32; ++j) { e[j] = rowmax - e[j]; m2 = fmaxf(m2, e[j]); }
    float s = 0.0f;
#pragma unroll
    for (int j = 0; j < 32; ++j) { e[j] = expf(e[j] - m2); s += e[j]; }
    const float scale = gamma[0] / s;   // fold gamma into attention
#pragma unroll
    for (int j = 0; j < 32; ++j) A[b * 1024 + i * 32 + j] = e[j] * scale;
}

// ---------------------------------------------------------------------------
// Pass 2: out[b] = A'[b] * q[b] + x[b], 16-column D-strip per wave.
// grid = (8192, B), block = 256 (8 waves). 16 WMMAs per wave (8 K-chunks x
// two 16-row halves), x-add fused on the C-fragment store.
// ---------------------------------------------------------------------------
__global__ void out_kernel(const float* __restrict__ x,
                           const float* __restrict__ A,
                           float* __restrict__ out) {
    const int b    = blockIdx.y;
    const int lane = threadIdx.x & 31;
    const int wave = threadIdx.x >> 5;

    const long long d0 = ((long long)blockIdx.x * 8 + wave) * 16;
    const float* q  = x   + (long long)b * N_ROWS * D_TOT;
    float*       o  = out + (long long)b * N_ROWS * D_TOT;
    const float* Ab = A   + b * 1024;

    const int half = lane >> 4;
    const int lm   = lane & 15;

    v8f c0 = {}, c1 = {};
#pragma unroll
    for (int kc = 0; kc < 8; ++kc) {
        const int m0 = kc * 4 + half * 2;
        // B fragment: q[m][d0+n] for this K-chunk (coalesced per half-wave)
        v2f bf;
        bf[0] = q[(long long)m0       * D_TOT + d0 + lm];
        bf[1] = q[(long long)(m0 + 1) * D_TOT + d0 + lm];
        // A fragments: attention rows 0..15 and 16..31 (4 KB table, cache-hot)
        v2f a0, a1;
        a0[0] = Ab[lm * 32 + m0];
        a0[1] = Ab[lm * 32 + m0 + 1];
        a1[0] = Ab[(16 + lm) * 32 + m0];
        a1[1] = Ab[(16 + lm) * 32 + m0 + 1];
        c0 = __builtin_amdgcn_wmma_f32_16x16x4_f32(false, a0, false, bf, (short)0, c0, false, false);
        c1 = __builtin_amdgcn_wmma_f32_16x16x4_f32(false, a1, false, bf, (short)0, c1, false, false);
    }

    const int rbase = half * 8;
#pragma unroll
    for (int r = 0; r < 8; ++r) {
        const long long i0 = (long long)(rbase + r)      * D_TOT + d0 + lm;
        const long long i1 = (long long)(16 + rbase + r) * D_TOT + d0 + lm;
        o[i0] = c0[r] + q[i0];   // attention already carries gamma
        o[i1] = c1[r] + q[i1];
    }
}

// ---------------------------------------------------------------------------
extern "C" void kernel_launch(void* const* d_in, const int* in_sizes, int n_in,
                              void* d_out, int out_size, void* d_ws, size_t ws_size,
                              hipStream_t stream) {
    (void)in_sizes; (void)n_in; (void)out_size; (void)ws_size;

    const float* x     = (const float*)d_in[0];
    const float* gamma = (const float*)d_in[1];
    float*       out   = (float*)d_out;

    float* G   = (float*)d_ws;   // 2 * 32 * 32 floats
    float* Att = G + 2048;       // 2 * 32 * 32 floats

    zero_kernel<<<1, 256, 0, stream>>>(G, 2048);

    dim3 g1(512, 2);             // D / (8 waves * KPW) = 512 per batch
    gram_kernel<<<g1, 256, 0, stream>>>(x, G);

    attn_kernel<<<1, 64, 0, stream>>>(G, gamma, Att);

    dim3 g2(8192, 2);            // D / (8 waves * 16 cols) = 8192 per batch
    out_kernel<<<g2, 256, 0, stream>>>(x, Att, out);
}